// BoundaryLoss_61203283968394
// MI455X (gfx1250) — compile-verified
//
#include <hip/hip_runtime.h>

// ---------------------------------------------------------------------------
// BoundaryLoss for MI455X (gfx1250, wave32).
// predict, target: [32,1,512,512] f32. Output: scalar mean of weighted BCE.
// Bandwidth-bound (~71MB traffic -> ~3us @ 23.3TB/s). Async memory->LDS
// staging of the erosion halo, float4 global loads, raw-transcendental
// softplus, WMMA-f32 wave reduction, deterministic 2-pass sum.
// ---------------------------------------------------------------------------

typedef __attribute__((ext_vector_type(2))) float v2f;
typedef __attribute__((ext_vector_type(8))) float v8f;
typedef __attribute__((ext_vector_type(4))) int   v4i;

// Address-space-qualified 16-byte vector pointers for the async LDS builtin.
typedef __attribute__((address_space(1))) v4i gv4i;   // global
typedef __attribute__((address_space(3))) v4i lv4i;   // LDS

#define BATCH     32
#define IMG_H     512
#define IMG_W     512
#define ROWS      16                     // strip height per workgroup
#define NTHREADS  256                    // 8 waves of 32
#define NWAVES    (NTHREADS / 32)
#define NBLOCKS   (BATCH * (IMG_H / ROWS))   // 1024
#define NTOTAL    (BATCH * IMG_H * IMG_W)    // 8388608
#define BW_WEIGHT 5.0f

#define HAS_ASYNC_LDS __has_builtin(__builtin_amdgcn_global_load_async_to_lds_b128)

// ln(1 + exp(-a)) for a >= 0. 1+e is in [1,2] -> raw v_log_f32 (log2) is
// accurate there with no range fixup; v_exp_f32 is exp2.
__device__ __forceinline__ float softplus_neg(float a) {
#if __has_builtin(__builtin_amdgcn_exp2f) && __has_builtin(__builtin_amdgcn_logf)
    const float e = __builtin_amdgcn_exp2f(-a * 1.44269504088896340736f); // exp(-a)
    return __builtin_amdgcn_logf(1.0f + e) * 0.69314718055994530942f;     // ln(1+e)
#else
    return __logf(1.0f + __expf(-a));
#endif
}

// Exact f32 wave(32) reduction using V_WMMA_F32_16X16X4_F32.
// Each lane contributes two partials (p,q) -> A is a full 16x4 f32 matrix
// (lane l<16: A[l][0..1] = {p,q}; lane l>=16: A[l-16][2..3] = {p,q}).
// B = ones(4x16) => D[m][n] = p_m + q_m + p_{m+16} + q_{m+16} (indep. of n).
// Summing the 8 D VGPRs gives sum(r_0..7) in lanes 0-15 and sum(r_8..15) in
// lanes 16-31; a half-wave swap + add yields the full wave total in all lanes.
__device__ __forceinline__ float wave_reduce_wmma(float p, float q) {
    v2f a; a[0] = p;    a[1] = q;
    v2f b; b[0] = 1.0f; b[1] = 1.0f;
    v8f c = {0.f, 0.f, 0.f, 0.f, 0.f, 0.f, 0.f, 0.f};
    v8f d = __builtin_amdgcn_wmma_f32_16x16x4_f32(
        /*neg_a=*/false, a, /*neg_b=*/false, b,
        /*c_mod=*/(short)0, c, /*reuse_a=*/false, /*reuse_b=*/false);
    float s = ((d[0] + d[1]) + (d[2] + d[3])) + ((d[4] + d[5]) + (d[6] + d[7]));
    s += __shfl_xor(s, 16, 32);          // combine half-waves
    return s;
}

__global__ __launch_bounds__(NTHREADS)
void BoundaryLoss_partial_kernel(const float* __restrict__ predict,
                                 const float* __restrict__ target,
                                 float* __restrict__ partial) {
    __shared__ float tile[(ROWS + 2) * IMG_W];   // 36.9 KB: strip + halo rows
    __shared__ float wsum[NWAVES];

    const int tid   = threadIdx.x;
    const int bid   = blockIdx.x;
    const int img   = bid / (IMG_H / ROWS);
    const int strip = bid % (IMG_H / ROWS);
    const int r0    = strip * ROWS;
    const size_t img_base = (size_t)img * IMG_H * IMG_W;

    // ---- Stage target rows [r0-1, r0+ROWS] into LDS; zero out-of-image rows
    // (matches scipy erosion border_value=0 / jnp.pad zeros). Rows are
    // wave-uniform (128 float4/row, 32-lane waves), so EXEC is full on the
    // async path.
    const int nvec = (ROWS + 2) * IMG_W / 4;     // 2304 float4
    for (int i = tid; i < nvec; i += NTHREADS) {
        const int tr = i >> 7;                   // 128 float4 per row
        const int vc = (i & 127) << 2;
        const int gy = r0 - 1 + tr;
        float* lp = &tile[tr * IMG_W + vc];
        if (gy >= 0 && gy < IMG_H) {
            const float* gp = &target[img_base + (size_t)gy * IMG_W + vc];
#if HAS_ASYNC_LDS
            __builtin_amdgcn_global_load_async_to_lds_b128(
                (gv4i*)gp, (lv4i*)lp, /*offset=*/0, /*cpol=*/0);
#else
            *reinterpret_cast<float4*>(lp) =
                *reinterpret_cast<const float4*>(gp);
#endif
        } else {
            *reinterpret_cast<float4*>(lp) = make_float4(0.f, 0.f, 0.f, 0.f);
        }
    }
#if HAS_ASYNC_LDS
    // Drain this wave's async memory->LDS copies before the barrier.
#if __has_builtin(__builtin_amdgcn_s_wait_asynccnt)
    __builtin_amdgcn_s_wait_asynccnt(0);
#else
    asm volatile("s_wait_asynccnt 0x0" ::: "memory");
#endif
#endif
    __syncthreads();

    // ---- Main compute: ROWS x 128 float4 = 2048 vectors, 8 per thread.
    float accp = 0.f, accq = 0.f;                // dual accumulators (ILP + WMMA A)
#pragma unroll
    for (int it = 0; it < (ROWS * IMG_W / 4) / NTHREADS; ++it) {
        const int vidx = it * NTHREADS + tid;
        const int j    = vidx >> 7;              // strip row 0..ROWS-1
        const int vc   = (vidx & 127) << 2;      // column of float4
        const int trow = j + 1;                  // tile row (halo offset)

        const float4 cv = *reinterpret_cast<const float4*>(&tile[trow * IMG_W + vc]);
        const float4 uv = *reinterpret_cast<const float4*>(&tile[(trow - 1) * IMG_W + vc]);
        const float4 dv = *reinterpret_cast<const float4*>(&tile[(trow + 1) * IMG_W + vc]);
        const float lfs = (vc == 0)         ? 0.f : tile[trow * IMG_W + vc - 1];
        const float rts = (vc + 4 == IMG_W) ? 0.f : tile[trow * IMG_W + vc + 4];

        const float4 pv = *reinterpret_cast<const float4*>(
                              &predict[img_base + (size_t)(r0 + j) * IMG_W + vc]);

        const float ca[4] = {cv.x, cv.y, cv.z, cv.w};
        const float ua[4] = {uv.x, uv.y, uv.z, uv.w};
        const float da[4] = {dv.x, dv.y, dv.z, dv.w};
        const float xa[4] = {pv.x, pv.y, pv.z, pv.w};
#pragma unroll
        for (int k = 0; k < 4; ++k) {
            const float lf = (k == 0) ? lfs : ca[k - 1];
            const float rt = (k == 3) ? rts : ca[k + 1];
            const float er = fminf(ca[k], fminf(fminf(ua[k], da[k]), fminf(lf, rt)));
            const float wgt = 1.0f + BW_WEIGHT * (ca[k] - er);
            const float x  = xa[k];
            // stable BCE-with-logits: max(x,0) - x*t + log1p(exp(-|x|))
            const float pl = fmaxf(x, 0.f) - x * ca[k] + softplus_neg(fabsf(x));
            if (k & 1) accq = fmaf(pl, wgt, accq);
            else       accp = fmaf(pl, wgt, accp);
        }
    }

    // ---- Wave reduction on the matrix pipe, then cross-wave via LDS.
    const float wtot = wave_reduce_wmma(accp, accq);
    const int lane = tid & 31, wv = tid >> 5;
    if (lane == 0) wsum[wv] = wtot;
    __syncthreads();
    if (tid == 0) {
        float s = 0.f;
#pragma unroll
        for (int w = 0; w < NWAVES; ++w) s += wsum[w];
        partial[bid] = s;
    }
}

__global__ __launch_bounds__(NTHREADS)
void BoundaryLoss_final_kernel(const float* __restrict__ partial,
                               float* __restrict__ out) {
    __shared__ float wsum[NWAVES];
    const int tid = threadIdx.x;
    float p = 0.f, q = 0.f;
    // NBLOCKS (1024) is a multiple of 2*NTHREADS, so both reads are in range.
    for (int i = tid; i < NBLOCKS; i += 2 * NTHREADS) {
        p += partial[i];
        q += partial[i + NTHREADS];
    }
    const float wtot = wave_reduce_wmma(p, q);
    const int lane = tid & 31, wv = tid >> 5;
    if (lane == 0) wsum[wv] = wtot;
    __syncthreads();
    if (tid == 0) {
        float s = 0.f;
#pragma unroll
        for (int w = 0; w < NWAVES; ++w) s += wsum[w];
        out[0] = s * (1.0f / (float)NTOTAL);
    }
}

extern "C" void kernel_launch(void* const* d_in, const int* in_sizes, int n_in,
                              void* d_out, int out_size, void* d_ws, size_t ws_size,
                              hipStream_t stream) {
    (void)in_sizes; (void)n_in; (void)out_size; (void)ws_size;
    const float* predict = (const float*)d_in[0];
    const float* target  = (const float*)d_in[1];
    float* out     = (float*)d_out;
    float* partial = (float*)d_ws;               // 1024 floats = 4 KB scratch

    BoundaryLoss_partial_kernel<<<NBLOCKS, NTHREADS, 0, stream>>>(predict, target, partial);
    BoundaryLoss_final_kernel<<<1, NTHREADS, 0, stream>>>(partial, out);
}